// LinearInterpolationModule_50921132261445
// MI455X (gfx1250) — compile-verified
//
#include <hip/hip_runtime.h>

// Problem constants (from reference): y: [BATCH, NPTS] f32, x_new: [MQ] f32,
// out: [BATCH, MQ] f32, uniform grid x_points = linspace(0,1,NPTS).
#define NPTS 2048
#define MQ   4096
#define TPB  256      // 8 wave32s per block
#define RPB  8        // batch rows per block

typedef float v4f __attribute__((ext_vector_type(4)));
typedef int   v4i __attribute__((ext_vector_type(4)));
typedef int   v4i_b __attribute__((vector_size(16)));   // matches builtin param

// ---- CDNA5 async global->LDS path (ASYNCcnt) ------------------------------
#if __has_builtin(__builtin_amdgcn_global_load_async_to_lds_b128)
#define HAVE_ASYNC_LDS 1
#endif

#if __has_builtin(__builtin_amdgcn_s_wait_asynccnt)
#define WAIT_ASYNC(n) __builtin_amdgcn_s_wait_asynccnt(n)
#elif defined(__AMDGCN__)
#define WAIT_ASYNC(n) asm volatile("s_wait_asynccnt %0" ::"n"(n) : "memory")
#else
#define WAIT_ASYNC(n) do {} while (0)
#endif

__device__ __forceinline__ void copy16_g2l(const float* g, float* l) {
#if defined(HAVE_ASYNC_LDS)
  __builtin_amdgcn_global_load_async_to_lds_b128(
      (__attribute__((address_space(1))) v4i_b*)(g),
      (__attribute__((address_space(3))) v4i_b*)(l),
      /*offset=*/0, /*cpol=*/0);
#else
  *(v4f*)l = *(const v4f*)g;   // sync fallback
#endif
}

// Issue one 8KB row copy: 512 x b128, 256 threads x 2 instructions each
// -> exactly 2 outstanding ASYNCcnt entries per wave per row.
__device__ __forceinline__ void load_row(const float* __restrict__ g,
                                         float* l, int tid) {
  copy16_g2l(g + tid * 4,          l + tid * 4);
  copy16_g2l(g + (tid + TPB) * 4,  l + (tid + TPB) * 4);
}

__global__ __launch_bounds__(TPB)
void interp1d_kernel(const float* __restrict__ y,
                     const float* __restrict__ xn,
                     float* __restrict__ out,
                     int batch) {
  __shared__ int   s_idx[MQ];              // 16 KB
  __shared__ float s_w[MQ];                // 16 KB
  __shared__ float s_row[2][NPTS];         // 2 x 8 KB double buffer

  const int tid  = threadIdx.x;
  const int row0 = blockIdx.x * RPB;

  // Prefetch first row asynchronously; overlaps with index/weight setup.
  if (row0 < batch) load_row(y + (size_t)row0 * NPTS, s_row[0], tid);

  // Phase 1: per-query segment index + weight (uniform grid searchsorted).
  for (int j = tid; j < MQ; j += TPB) {
    float x = xn[j];
    int i = (int)(x * (float)(NPTS - 1));
    i = min(max(i, 0), NPTS - 2);
    // fix up against the exact grid values i/(NPTS-1) (1-ulp rounding guard)
    float x1 = (float)i / (float)(NPTS - 1);
    if (x < x1) {
      i = i - 1;
    } else if (x >= (float)(i + 1) / (float)(NPTS - 1)) {
      i = i + 1;
    }
    i = min(max(i, 0), NPTS - 2);
    x1        = (float)i / (float)(NPTS - 1);
    float x2  = (float)(i + 1) / (float)(NPTS - 1);
    s_idx[j] = i;
    s_w[j]   = (x - x1) / (x2 - x1);
  }

  int buf = 0;
  for (int r = 0; r < RPB; ++r) {
    const int row = row0 + r;
    if (row >= batch) break;

    // Prefetch next row into the other buffer, then wait so that only the
    // prefetch (<=2 per wave) is still outstanding -> current row is in LDS.
    const bool pf = (r + 1 < RPB) && (row + 1 < batch);
    if (pf) {
      load_row(y + (size_t)(row + 1) * NPTS, s_row[buf ^ 1], tid);
      WAIT_ASYNC(2);
    } else {
      WAIT_ASYNC(0);
    }
    __syncthreads();   // all waves' portions of this row visible (+ phase 1)

    const float* rp = s_row[buf];
    float* op = out + (size_t)row * MQ;

#pragma unroll
    for (int t = 0; t < MQ / (TPB * 4); ++t) {          // 4 iterations
      const int j = (tid + t * TPB) * 4;
      v4i iv = *(const v4i*)&s_idx[j];
      v4f wv = *(const v4f*)&s_w[j];
      v4f res;
      {
        float a = rp[iv.x], b = rp[iv.x + 1];
        res.x = fmaf(wv.x, b - a, a);
      }
      {
        float a = rp[iv.y], b = rp[iv.y + 1];
        res.y = fmaf(wv.y, b - a, a);
      }
      {
        float a = rp[iv.z], b = rp[iv.z + 1];
        res.z = fmaf(wv.z, b - a, a);
      }
      {
        float a = rp[iv.w], b = rp[iv.w + 1];
        res.w = fmaf(wv.w, b - a, a);
      }
      __builtin_nontemporal_store(res, (v4f*)&op[j]);   // streamed, write-once
    }

    __syncthreads();   // everyone done reading s_row[buf] before it is reused
    buf ^= 1;
  }
}

extern "C" void kernel_launch(void* const* d_in, const int* in_sizes, int n_in,
                              void* d_out, int out_size, void* d_ws, size_t ws_size,
                              hipStream_t stream) {
  (void)n_in; (void)out_size; (void)d_ws; (void)ws_size;
  const float* y  = (const float*)d_in[0];   // [batch, NPTS]
  const float* xn = (const float*)d_in[1];   // [MQ]
  float* out = (float*)d_out;                // [batch, MQ]
  const int batch = in_sizes[0] / NPTS;
  const int grid  = (batch + RPB - 1) / RPB;
  hipLaunchKernelGGL(interp1d_kernel, dim3(grid), dim3(TPB), 0, stream,
                     y, xn, out, batch);
}